// GTFShallowPLRNN_60971355734108
// MI455X (gfx1250) — compile-verified
//
#include <hip/hip_runtime.h>
#include <hip/hip_bf16.h>

typedef float v2f __attribute__((ext_vector_type(2)));
typedef float v8f __attribute__((ext_vector_type(8)));

// D = A(16x4) * B(4x16) + C, fp32 exact WMMA (CDNA5 V_WMMA_F32_16X16X4_F32)
__device__ __forceinline__ v8f wmma4(v2f a, v2f b, v8f c) {
    return __builtin_amdgcn_wmma_f32_16x16x4_f32(
        /*neg_a=*/false, a, /*neg_b=*/false, b,
        /*c_mod=*/(short)0, c, /*reuse_a=*/false, /*reuse_b=*/false);
}

// ---------------------------------------------------------------------------
// Kernel 1: conv1 (stride 10, k=50) + bn1 + relu as implicit-GEMM
// M = 512*246 = 125952, N = 32, K = 600 = 12 ic * 50 kk.
// One wave computes a 16x32 tile (both oc halves share A fragments).
// K consumed as: per-ic kk groups [0,48) + 6 cross-ic remainder groups.
// y1 layout: (B, 32, 246)
// ---------------------------------------------------------------------------
__global__ __launch_bounds__(256)
void conv1_kernel(const float* __restrict__ x, const float* __restrict__ w,
                  const float* __restrict__ cb, const float* __restrict__ g,
                  const float* __restrict__ bb, const float* __restrict__ mean,
                  const float* __restrict__ var, float* __restrict__ y1)
{
    int lane = threadIdx.x & 31;
    int wv   = threadIdx.x >> 5;
    int tm   = blockIdx.x * 8 + wv;       // 0..7871
    int n0   = lane & 15;
    int hi   = lane >> 4;
    int khalf = hi * 2;

    int m = tm * 16 + n0;                 // A row (same for both lane halves)
    int b = m / 246;
    int t = m - b * 246;
    const float* xrow = x + b * 30000 + t * 10;   // 12*2500 = 30000
    const float* wnA = w + n0 * 600;              // conv1_w flat: oc*600 + ic*50 + kk
    const float* wnB = w + (n0 + 16) * 600;

    v8f acc0 = {}, acc1 = {};
    for (int ic = 0; ic < 12; ++ic) {
        const float* xi = xrow + ic * 2500;
        const float* wA = wnA + ic * 50;
        const float* wB = wnB + ic * 50;
#pragma unroll
        for (int kk = 0; kk < 48; kk += 4) {
            int ka = kk + khalf;
            v2f a   = { xi[ka], xi[ka + 1] };
            v2f bf0 = { wA[ka], wA[ka + 1] };
            v2f bf1 = { wB[ka], wB[ka + 1] };
            acc0 = wmma4(a, bf0, acc0);
            acc1 = wmma4(a, bf1, acc1);
        }
    }
    // remainder taps kk=48,49 for all 12 ic: 6 groups, each pairing 2 ics.
#pragma unroll
    for (int j = 0; j < 6; ++j) {
        int ic = 2 * j + hi;              // lane-half selects ic within the pair
        v2f a   = { xrow[ic * 2500 + 48], xrow[ic * 2500 + 49] };
        v2f bf0 = { wnA[ic * 50 + 48],    wnA[ic * 50 + 49] };
        v2f bf1 = { wnB[ic * 50 + 48],    wnB[ic * 50 + 49] };
        acc0 = wmma4(a, bf0, acc0);
        acc1 = wmma4(a, bf1, acc1);
    }

    float s0  = g[n0] * rsqrtf(var[n0] + 1e-5f);
    float sh0 = (cb[n0] - mean[n0]) * s0 + bb[n0];
    int n1 = n0 + 16;
    float s1  = g[n1] * rsqrtf(var[n1] + 1e-5f);
    float sh1 = (cb[n1] - mean[n1]) * s1 + bb[n1];

    int rbase = hi * 8;
    int mo = tm * 16 + rbase;
    int bo = mo / 246;
    int to = mo - bo * 246;
#pragma unroll
    for (int v = 0; v < 8; ++v) {
        float* yb = y1 + bo * 7872 + to;
        yb[n0 * 246] = fmaxf(acc0[v] * s0 + sh0, 0.f);
        yb[n1 * 246] = fmaxf(acc1[v] * s1 + sh1, 0.f);
        if (++to == 246) { to = 0; ++bo; }
    }
}

// ---------------------------------------------------------------------------
// Kernel 2: conv2 (stride 5, k=25) + bn2 + relu as implicit-GEMM
// M = 512*45 = 23040, N = 32, K = 800 = 32 ic * 25 kk.
// One wave computes a 16x32 tile. K: per-ic kk groups [0,24) + 8 cross-ic
// remainder groups (kk=24, 4 ics each). Output stored transposed (B*45, 32).
// ---------------------------------------------------------------------------
__global__ __launch_bounds__(256)
void conv2_kernel(const float* __restrict__ y1, const float* __restrict__ w,
                  const float* __restrict__ cb, const float* __restrict__ g,
                  const float* __restrict__ bb, const float* __restrict__ mean,
                  const float* __restrict__ var, float* __restrict__ y2t)
{
    int lane = threadIdx.x & 31;
    int wv   = threadIdx.x >> 5;
    int tm   = blockIdx.x * 8 + wv;       // 0..1439
    int n0   = lane & 15;
    int hi   = lane >> 4;
    int khalf = hi * 2;

    int m  = tm * 16 + n0;
    int b  = m / 45;
    int t2 = m - b * 45;
    const float* arow = y1 + b * 7872 + t2 * 5;   // + ic*246 + kk
    const float* wnA = w + n0 * 800;              // conv2_w flat: oc*800 + ic*25 + kk
    const float* wnB = w + (n0 + 16) * 800;

    v8f acc0 = {}, acc1 = {};
    for (int ic = 0; ic < 32; ++ic) {
        const float* ai = arow + ic * 246;
        const float* wA = wnA + ic * 25;
        const float* wB = wnB + ic * 25;
#pragma unroll
        for (int kk = 0; kk < 24; kk += 4) {
            int ka = kk + khalf;
            v2f a   = { ai[ka], ai[ka + 1] };
            v2f bf0 = { wA[ka], wA[ka + 1] };
            v2f bf1 = { wB[ka], wB[ka + 1] };
            acc0 = wmma4(a, bf0, acc0);
            acc1 = wmma4(a, bf1, acc1);
        }
    }
    // remainder tap kk=24 for all 32 ic: 8 groups of 4 consecutive ics.
#pragma unroll
    for (int j = 0; j < 8; ++j) {
        int ic0 = 4 * j + 2 * hi;         // lane-half selects ic pair in the group
        int ic1 = ic0 + 1;
        v2f a   = { arow[ic0 * 246 + 24], arow[ic1 * 246 + 24] };
        v2f bf0 = { wnA[ic0 * 25 + 24],   wnA[ic1 * 25 + 24] };
        v2f bf1 = { wnB[ic0 * 25 + 24],   wnB[ic1 * 25 + 24] };
        acc0 = wmma4(a, bf0, acc0);
        acc1 = wmma4(a, bf1, acc1);
    }

    float s0  = g[n0] * rsqrtf(var[n0] + 1e-5f);
    float sh0 = (cb[n0] - mean[n0]) * s0 + bb[n0];
    int n1 = n0 + 16;
    float s1  = g[n1] * rsqrtf(var[n1] + 1e-5f);
    float sh1 = (cb[n1] - mean[n1]) * s1 + bb[n1];

    int rbase = hi * 8;
#pragma unroll
    for (int v = 0; v < 8; ++v) {
        int mo = tm * 16 + rbase + v;
        y2t[mo * 32 + n0] = fmaxf(acc0[v] * s0 + sh0, 0.f);
        y2t[mo * 32 + n1] = fmaxf(acc1[v] * s1 + sh1, 0.f);
    }
}

// ---------------------------------------------------------------------------
// Kernel 3: projection GEMM (M=23040, N=32, K=32) + LayerNorm(32).
// Writes xt in time-major layout (45, 512, 32) for the recurrence.
// ---------------------------------------------------------------------------
__global__ __launch_bounds__(256)
void proj_ln_kernel(const float* __restrict__ y2t, const float* __restrict__ pw,
                    const float* __restrict__ pb, const float* __restrict__ lg,
                    const float* __restrict__ lb, float* __restrict__ xt)
{
    __shared__ float tile_s[8][16][32];
    int lane = threadIdx.x & 31;
    int wv   = threadIdx.x >> 5;
    int tm   = blockIdx.x * 8 + wv;       // 0..1439
    int n0   = lane & 15;
    int hi   = lane >> 4;
    int khalf = hi * 2, rbase = hi * 8;

    const float* arow = y2t + (tm * 16 + n0) * 32;
    v8f c0 = {}, c1 = {};
#pragma unroll
    for (int k0 = 0; k0 < 32; k0 += 4) {
        int ka = k0 + khalf;
        v2f a  = { arow[ka], arow[ka + 1] };
        // B[f][d] = proj_w[d*32 + f]
        v2f b0 = { pw[n0 * 32 + ka],        pw[n0 * 32 + ka + 1] };
        v2f b1 = { pw[(n0 + 16) * 32 + ka], pw[(n0 + 16) * 32 + ka + 1] };
        c0 = wmma4(a, b0, c0);
        c1 = wmma4(a, b1, c1);
    }
#pragma unroll
    for (int v = 0; v < 8; ++v) {
        int r = rbase + v;
        tile_s[wv][r][n0]      = c0[v] + pb[n0];
        tile_s[wv][r][n0 + 16] = c1[v] + pb[n0 + 16];
    }
    __syncthreads();
    if (lane < 16) {
        float* hr = tile_s[wv][lane];
        float mu = 0.f;
        for (int d = 0; d < 32; ++d) mu += hr[d];
        mu *= (1.f / 32.f);
        float vv = 0.f;
        for (int d = 0; d < 32; ++d) { float df = hr[d] - mu; vv += df * df; }
        vv *= (1.f / 32.f);
        float inv = rsqrtf(vv + 1e-5f);
        int mg = tm * 16 + lane;
        int b = mg / 45, t = mg - b * 45;
        float* orow = xt + t * 16384 + b * 32;    // 512*32 = 16384
        for (int d = 0; d < 32; ++d)
            orow[d] = (hr[d] - mu) * inv * lg[d] + lb[d];
    }
}

// ---------------------------------------------------------------------------
// Kernel 4: 45-step PLRNN recurrence + pooling + output GEMM.
// Batch rows are independent: each wave owns 16 rows for all 45 steps.
// All GEMMs via fp32 WMMA; LN round-trips the 16x64 tile through LDS.
// h_new = h_pred + 0.9 * x_t  (algebraic form of the alpha blend).
// ---------------------------------------------------------------------------
__global__ __launch_bounds__(128)
void rnn_kernel(const float* __restrict__ xt, const float* __restrict__ Amat,
                const float* __restrict__ W1, const float* __restrict__ W2,
                const float* __restrict__ h1, const float* __restrict__ h2,
                const float* __restrict__ lg2, const float* __restrict__ lb2,
                const float* __restrict__ ow, const float* __restrict__ ob,
                float* __restrict__ out)
{
    __shared__ float W2T[32][64];   // W2T[k][n] = W2[n*32+k]   (h @ W2.T)
    __shared__ float W1T[64][32];   // W1T[k][n] = W1[n*64+k]   (hidden @ W1.T)
    __shared__ float AT [32][32];   // AT[k][n]  = A[n*32+k]    (h @ A.T)
    __shared__ float OWT[32][32];   // OWT[k][n] = out_w[n*32+k]
    __shared__ float h_s  [4][16][32];
    __shared__ float hid_s[4][16][64];

    int tid = threadIdx.x;
    for (int i = tid; i < 2048; i += 128) { int k = i >> 6, n = i & 63; W2T[k][n] = W2[n * 32 + k]; }
    for (int i = tid; i < 2048; i += 128) { int k = i >> 5, n = i & 31; W1T[k][n] = W1[n * 64 + k]; }
    for (int i = tid; i < 1024; i += 128) { int k = i >> 5, n = i & 31; AT[k][n] = Amat[n * 32 + k]; OWT[k][n] = ow[n * 32 + k]; }
    for (int i = tid; i < 2048; i += 128) ((float*)h_s)[i] = 0.f;   // h0 = 0
    __syncthreads();

    int lane = tid & 31, wv = tid >> 5;
    int n0 = lane & 15, hi = lane >> 4;
    int khalf = hi * 2, rbase = hi * 8;
    int m0 = (blockIdx.x * 4 + wv) * 16;   // first batch row of this wave

    v8f sum0 = {}, sum1 = {};

    for (int t = 0; t < 45; ++t) {
        // ---- hidden_pre = h @ W2.T  (16x64, K=32) ----
        v8f hc0 = {}, hc1 = {}, hc2 = {}, hc3 = {};
#pragma unroll
        for (int k0 = 0; k0 < 32; k0 += 4) {
            int ka = k0 + khalf;
            v2f a  = { h_s[wv][n0][ka], h_s[wv][n0][ka + 1] };
            v2f b0 = { W2T[ka][n0],      W2T[ka + 1][n0] };
            v2f b1 = { W2T[ka][16 + n0], W2T[ka + 1][16 + n0] };
            v2f b2 = { W2T[ka][32 + n0], W2T[ka + 1][32 + n0] };
            v2f b3 = { W2T[ka][48 + n0], W2T[ka + 1][48 + n0] };
            hc0 = wmma4(a, b0, hc0); hc1 = wmma4(a, b1, hc1);
            hc2 = wmma4(a, b2, hc2); hc3 = wmma4(a, b3, hc3);
        }
#pragma unroll
        for (int v = 0; v < 8; ++v) {
            int r = rbase + v;
            hid_s[wv][r][n0]      = fmaxf(hc0[v] + h2[n0],      0.f);
            hid_s[wv][r][16 + n0] = fmaxf(hc1[v] + h2[16 + n0], 0.f);
            hid_s[wv][r][32 + n0] = fmaxf(hc2[v] + h2[32 + n0], 0.f);
            hid_s[wv][r][48 + n0] = fmaxf(hc3[v] + h2[48 + n0], 0.f);
        }
        __syncthreads();
        // ---- LayerNorm over 64 per row ----
        if (lane < 16) {
            float* hr = hid_s[wv][lane];
            float mu = 0.f;
            for (int d = 0; d < 64; ++d) mu += hr[d];
            mu *= (1.f / 64.f);
            float vv = 0.f;
            for (int d = 0; d < 64; ++d) { float df = hr[d] - mu; vv += df * df; }
            vv *= (1.f / 64.f);
            float inv = rsqrtf(vv + 1e-5f);
            for (int d = 0; d < 64; ++d) hr[d] = (hr[d] - mu) * inv * lg2[d] + lb2[d];
        }
        __syncthreads();
        // ---- h_pred = h @ A.T (K=32)  +  hidden @ W1.T (K=64), same accumulator ----
        v8f c0 = {}, c1 = {};
#pragma unroll
        for (int k0 = 0; k0 < 32; k0 += 4) {
            int ka = k0 + khalf;
            v2f a  = { h_s[wv][n0][ka], h_s[wv][n0][ka + 1] };
            v2f b0 = { AT[ka][n0],      AT[ka + 1][n0] };
            v2f b1 = { AT[ka][16 + n0], AT[ka + 1][16 + n0] };
            c0 = wmma4(a, b0, c0); c1 = wmma4(a, b1, c1);
        }
#pragma unroll
        for (int k0 = 0; k0 < 64; k0 += 4) {
            int ka = k0 + khalf;
            v2f a  = { hid_s[wv][n0][ka], hid_s[wv][n0][ka + 1] };
            v2f b0 = { W1T[ka][n0],      W1T[ka + 1][n0] };
            v2f b1 = { W1T[ka][16 + n0], W1T[ka + 1][16 + n0] };
            c0 = wmma4(a, b0, c0); c1 = wmma4(a, b1, c1);
        }
        __syncthreads();
        // ---- h_new = h_pred + h1 + 0.9 * x_t ; accumulate pooled sum ----
        const float* xrow = xt + t * 16384 + m0 * 32;
#pragma unroll
        for (int v = 0; v < 8; ++v) {
            int r = rbase + v;
            float hn0 = c0[v] + h1[n0]      + 0.9f * xrow[r * 32 + n0];
            float hn1 = c1[v] + h1[16 + n0] + 0.9f * xrow[r * 32 + 16 + n0];
            h_s[wv][r][n0]      = hn0;
            h_s[wv][r][16 + n0] = hn1;
            sum0[v] += hn0; sum1[v] += hn1;
        }
        __syncthreads();
    }

    // ---- pooled = mean_t(h);  out = pooled @ out_w.T + out_b ----
#pragma unroll
    for (int v = 0; v < 8; ++v) {
        int r = rbase + v;
        h_s[wv][r][n0]      = sum0[v] * (1.f / 45.f);
        h_s[wv][r][16 + n0] = sum1[v] * (1.f / 45.f);
    }
    __syncthreads();
    v8f o0 = {}, o1 = {};
#pragma unroll
    for (int k0 = 0; k0 < 32; k0 += 4) {
        int ka = k0 + khalf;
        v2f a  = { h_s[wv][n0][ka], h_s[wv][n0][ka + 1] };
        v2f b0 = { OWT[ka][n0],      OWT[ka + 1][n0] };
        v2f b1 = { OWT[ka][16 + n0], OWT[ka + 1][16 + n0] };
        o0 = wmma4(a, b0, o0); o1 = wmma4(a, b1, o1);
    }
#pragma unroll
    for (int v = 0; v < 8; ++v) {
        int r = rbase + v;
        out[(m0 + r) * 32 + n0]      = o0[v] + ob[n0];
        out[(m0 + r) * 32 + 16 + n0] = o1[v] + ob[16 + n0];
    }
}

// ---------------------------------------------------------------------------
extern "C" void kernel_launch(void* const* d_in, const int* in_sizes, int n_in,
                              void* d_out, int out_size, void* d_ws, size_t ws_size,
                              hipStream_t stream)
{
    const float* x       = (const float*)d_in[0];
    const float* conv1_w = (const float*)d_in[1];
    const float* conv1_b = (const float*)d_in[2];
    const float* bn1_g   = (const float*)d_in[3];
    const float* bn1_b   = (const float*)d_in[4];
    const float* bn1_m   = (const float*)d_in[5];
    const float* bn1_v   = (const float*)d_in[6];
    const float* conv2_w = (const float*)d_in[7];
    const float* conv2_b = (const float*)d_in[8];
    const float* bn2_g   = (const float*)d_in[9];
    const float* bn2_b   = (const float*)d_in[10];
    const float* bn2_m   = (const float*)d_in[11];
    const float* bn2_v   = (const float*)d_in[12];
    const float* proj_w  = (const float*)d_in[13];
    const float* proj_b  = (const float*)d_in[14];
    const float* Amat    = (const float*)d_in[15];
    const float* W1      = (const float*)d_in[16];
    const float* W2      = (const float*)d_in[17];
    const float* h1      = (const float*)d_in[18];
    const float* h2      = (const float*)d_in[19];
    const float* ln1_g   = (const float*)d_in[20];
    const float* ln1_b   = (const float*)d_in[21];
    const float* ln2_g   = (const float*)d_in[22];
    const float* ln2_b   = (const float*)d_in[23];
    const float* out_w   = (const float*)d_in[24];
    const float* out_b   = (const float*)d_in[25];

    float* ws  = (float*)d_ws;
    float* y1  = ws;                       // 512*32*246 = 4,030,464 floats
    float* y2t = ws + 4030464;             // 23040*32   =   737,280 floats
    float* xtb = ws + 4030464 + 737280;    // 45*512*32  =   737,280 floats

    conv1_kernel  <<<984, 256, 0, stream>>>(x, conv1_w, conv1_b, bn1_g, bn1_b, bn1_m, bn1_v, y1);
    conv2_kernel  <<<180, 256, 0, stream>>>(y1, conv2_w, conv2_b, bn2_g, bn2_b, bn2_m, bn2_v, y2t);
    proj_ln_kernel<<<180, 256, 0, stream>>>(y2t, proj_w, proj_b, ln1_g, ln1_b, xtb);
    rnn_kernel    <<<8,   128, 0, stream>>>(xtb, Amat, W1, W2, h1, h2, ln2_g, ln2_b,
                                            out_w, out_b, (float*)d_out);
}